// FaceConv_13099650253565
// MI455X (gfx1250) — compile-verified
//
#include <hip/hip_runtime.h>

typedef float v2f __attribute__((ext_vector_type(2)));
typedef float v4f __attribute__((ext_vector_type(4)));
typedef float v8f __attribute__((ext_vector_type(8)));

constexpr int C_IN  = 256;
constexpr int K1    = 9;
constexpr int KTOT  = C_IN * K1;   // 2304 contraction length
constexpr int C_OUT = 256;
constexpr int TILE_M = 32;         // faces per block (2 WMMA M-subtiles)
constexpr int LDS_STRIDE = 260;    // 256 + 4-float skew: conflict-free column reads

// ---------------------------------------------------------------------------
// Pre-pass: Bt[n][k1*256 + c] = W[n][c][k1]  (k-dimension made contiguous)
// ---------------------------------------------------------------------------
__global__ void transpose_W_kernel(const float* __restrict__ W,
                                   float* __restrict__ Bt) {
    const int n = blockIdx.x;       // output channel
    const int c = threadIdx.x;      // input channel
    const float* w = W + ((size_t)n * C_IN + c) * K1;
    float* o = Bt + (size_t)n * KTOT + c;
#pragma unroll
    for (int k1 = 0; k1 < K1; ++k1)
        o[k1 * C_IN] = w[k1];       // coalesced across c per k1
}

// ---------------------------------------------------------------------------
// Main kernel: gathered GEMM [M x 2304] * [2304 x 256] via f32 WMMA 16x16x4.
// Block = 256 threads (8 wave32); block owns 32 faces (2 M-subtiles of 16);
// each wave owns two 16-wide N-tiles (8 waves x 32 = 256 output channels).
// Inner 4-K step: 2 ds_load_b64 (A) + 2 global_load_b64 (B, L2) + 4 WMMA.
// A-chunks are register-prefetched one neighbor slot ahead (software pipeline).
// WT=true: B is the k-contiguous Bt layout; WT=false: strided reads of raw W.
// ---------------------------------------------------------------------------
template <bool WT>
__global__ __launch_bounds__(256)
void faceconv_wmma_kernel(const float* __restrict__ x,
                          const int*   __restrict__ nb,
                          const float* __restrict__ B,
                          const float* __restrict__ bias,
                          float*       __restrict__ out,
                          int n_faces) {
    __shared__ __align__(16) float Alds[TILE_M * LDS_STRIDE];
    __shared__ int nbr[TILE_M * K1];            // 32 x 9 resolved gather rows

    const int tid = threadIdx.x;
    const int m0  = blockIdx.x * TILE_M;

    // Resolve neighbor indices once (pad sentinel: idx >= n_faces -> -1 -> zeros).
    for (int i = tid; i < TILE_M * K1; i += 256) {
        const int idx = nb[(size_t)m0 * K1 + i];
        nbr[i] = (idx < n_faces) ? idx : -1;
    }
    __syncthreads();

    // Staging map: 8 threads per face-row, 32 floats (8 x b128) per thread.
    const int srow = tid >> 3;
    const int scol = (tid & 7) * 32;
    float* sdst = &Alds[srow * LDS_STRIDE + scol];

    v4f sreg[8];
    auto prefetch_chunk = [&](int k1) {
        const int row = nbr[srow * K1 + k1];
        if (row >= 0) {
            const float* src = x + (size_t)row * C_IN + scol;
#pragma unroll
            for (int j = 0; j < 8; ++j) sreg[j] = *(const v4f*)(src + 4 * j);
        } else {
#pragma unroll
            for (int j = 0; j < 8; ++j) sreg[j] = (v4f){0.f, 0.f, 0.f, 0.f};
        }
    };
    prefetch_chunk(0);

    const int lane = tid & 31;
    const int wave = tid >> 5;
    const int mrow = lane & 15;
    const int half = lane >> 4;     // 0: rel-K {0,1}, 1: rel-K {2,3}
    const int koff = half * 2;

    const int n0 = wave * 32 + (lane & 15);
    const int n1 = n0 + 16;

    // C/D 16x16 f32 layout: value depends only on N here -> broadcast bias.
    v8f acc[2][2];
    {
        const float b0 = bias[n0], b1 = bias[n1];
#pragma unroll
        for (int i = 0; i < 8; ++i) {
            acc[0][0][i] = b0; acc[0][1][i] = b1;
            acc[1][0][i] = b0; acc[1][1][i] = b1;
        }
    }

    // A-operand LDS bases for the two M-subtiles (ISA 16x4 f32 A layout).
    const float* aBase0 = &Alds[mrow * LDS_STRIDE + koff];
    const float* aBase1 = aBase0 + 16 * LDS_STRIDE;

    for (int k1 = 0; k1 < K1; ++k1) {
        __syncthreads();            // previous chunk's compute done
#pragma unroll
        for (int j = 0; j < 8; ++j) *(v4f*)(sdst + 4 * j) = sreg[j];
        __syncthreads();
        if (k1 + 1 < K1) prefetch_chunk(k1 + 1);   // overlap gather w/ WMMAs

        if (WT) {
            const float* b0p = B + (size_t)n0 * KTOT + k1 * C_IN + koff;
            const float* b1p = B + (size_t)n1 * KTOT + k1 * C_IN + koff;
#pragma unroll 4
            for (int kk = 0; kk < C_IN; kk += 4) {
                const v2f a0 = *(const v2f*)(aBase0 + kk);  // ds_load_b64
                const v2f a1 = *(const v2f*)(aBase1 + kk);
                const v2f b0 = *(const v2f*)(b0p + kk);     // global_load_b64
                const v2f b1 = *(const v2f*)(b1p + kk);
                acc[0][0] = __builtin_amdgcn_wmma_f32_16x16x4_f32(
                                false, a0, false, b0, (short)0, acc[0][0], false, false);
                acc[0][1] = __builtin_amdgcn_wmma_f32_16x16x4_f32(
                                false, a0, false, b1, (short)0, acc[0][1], false, false);
                acc[1][0] = __builtin_amdgcn_wmma_f32_16x16x4_f32(
                                false, a1, false, b0, (short)0, acc[1][0], false, false);
                acc[1][1] = __builtin_amdgcn_wmma_f32_16x16x4_f32(
                                false, a1, false, b1, (short)0, acc[1][1], false, false);
            }
        } else {
            // Raw W[n][c][k1]: element c of this chunk lives at base[c*9].
            const float* b0p = B + (size_t)n0 * KTOT + k1;
            const float* b1p = B + (size_t)n1 * KTOT + k1;
#pragma unroll 4
            for (int kk = 0; kk < C_IN; kk += 4) {
                const v2f a0 = *(const v2f*)(aBase0 + kk);
                const v2f a1 = *(const v2f*)(aBase1 + kk);
                const int c0 = kk + koff;
                const v2f b0 = { b0p[(size_t)c0 * K1], b0p[(size_t)(c0 + 1) * K1] };
                const v2f b1 = { b1p[(size_t)c0 * K1], b1p[(size_t)(c0 + 1) * K1] };
                acc[0][0] = __builtin_amdgcn_wmma_f32_16x16x4_f32(
                                false, a0, false, b0, (short)0, acc[0][0], false, false);
                acc[0][1] = __builtin_amdgcn_wmma_f32_16x16x4_f32(
                                false, a0, false, b1, (short)0, acc[0][1], false, false);
                acc[1][0] = __builtin_amdgcn_wmma_f32_16x16x4_f32(
                                false, a1, false, b0, (short)0, acc[1][0], false, false);
                acc[1][1] = __builtin_amdgcn_wmma_f32_16x16x4_f32(
                                false, a1, false, b1, (short)0, acc[1][1], false, false);
            }
        }
    }

    // D 16x16 f32 layout: VGPR r -> M = r + 8*half, N = lane&15.
#pragma unroll
    for (int msub = 0; msub < 2; ++msub) {
        float* o = out + (size_t)(m0 + msub * 16 + half * 8) * C_OUT;
#pragma unroll
        for (int r = 0; r < 8; ++r) {
            o[(size_t)r * C_OUT + n0] = acc[msub][0][r];
            o[(size_t)r * C_OUT + n1] = acc[msub][1][r];
        }
    }
}

// ---------------------------------------------------------------------------
// Inputs (setup_inputs order): x f32[65536*256], face_neighborhood int[65536*9],
// face_is_pad bool[66560] (unused: pads are exactly idx >= N_FACES),
// W f32[256*256*9], b f32[256], pad_size int[1] (unused).
// ---------------------------------------------------------------------------
extern "C" void kernel_launch(void* const* d_in, const int* in_sizes, int n_in,
                              void* d_out, int out_size, void* d_ws, size_t ws_size,
                              hipStream_t stream) {
    const float* x    = (const float*)d_in[0];
    const int*   nb   = (const int*)  d_in[1];
    const float* W    = (const float*)d_in[3];
    const float* bias = (const float*)d_in[4];
    float*       out  = (float*)d_out;

    const int n_faces = in_sizes[0] / C_IN;     // 65536
    const int nblocks = n_faces / TILE_M;       // 2048

    const size_t btBytes = (size_t)C_OUT * KTOT * sizeof(float);  // 2.25 MB
    if (ws_size >= btBytes) {
        float* Bt = (float*)d_ws;
        transpose_W_kernel<<<C_OUT, C_IN, 0, stream>>>(W, Bt);
        faceconv_wmma_kernel<true><<<nblocks, 256, 0, stream>>>(
            x, nb, Bt, bias, out, n_faces);
    } else {
        faceconv_wmma_kernel<false><<<nblocks, 256, 0, stream>>>(
            x, nb, W, bias, out, n_faces);
    }
}